// prms_marrmot_53781580480882
// MI455X (gfx1250) — compile-verified
//
#include <hip/hip_runtime.h>
#include <hip/hip_bf16.h>
#include <math.h>

// Problem constants (from reference)
#define NB_   2000
#define NT_   730
#define NMUL_ 8
#define NPRM_ 12
#define LENF_ 15

// Scan-kernel tiling
#define BASINS_PER_BLOCK 16
#define SCAN_BLOCK (BASINS_PER_BLOCK * NMUL_)     // 128 threads = 4 wave32
#define PROW 104                                   // padded LDS row (floats) -> conflict-free ds reads
#define SLAB_FLOATS (BASINS_PER_BLOCK * NPRM_ * NMUL_)  // 1536 floats / step / block
#define ASYNC_OPS 3                                // 128 lanes * 16B * 3 = 6144B = slab

// ---------------- CDNA5 async global->LDS helpers ----------------
typedef int v4i __attribute__((ext_vector_type(4)));
typedef __attribute__((address_space(1))) v4i gm_v4i;
typedef __attribute__((address_space(3))) v4i lds_v4i;

__device__ __forceinline__ void async_copy_b128(const void* g, void* l) {
#if defined(__has_builtin) && __has_builtin(__builtin_amdgcn_global_load_async_to_lds_b128)
  __builtin_amdgcn_global_load_async_to_lds_b128((gm_v4i*)g, (lds_v4i*)l, 0, 0);
#else
  unsigned lofs = (unsigned)(size_t)(__attribute__((address_space(3))) void*)l;
  asm volatile("global_load_async_to_lds_b128 %0, %1, off"
               :: "v"(lofs), "v"(g) : "memory");
#endif
}

template <int N>
__device__ __forceinline__ void wait_asynccnt() {
#if defined(__has_builtin) && __has_builtin(__builtin_amdgcn_s_wait_asynccnt)
  __builtin_amdgcn_s_wait_asynccnt(N);
#else
  asm volatile("s_wait_asynccnt %0" :: "i"(N) : "memory");
#endif
}

// sum over aligned groups of 8 lanes (wave32, EXEC all ones here)
__device__ __forceinline__ float grp8_sum(float x) {
  x += __int_as_float(__builtin_amdgcn_ds_swizzle(__float_as_int(x), 0x041F)); // xor 1
  x += __int_as_float(__builtin_amdgcn_ds_swizzle(__float_as_int(x), 0x081F)); // xor 2
  x += __int_as_float(__builtin_amdgcn_ds_swizzle(__float_as_int(x), 0x101F)); // xor 4
  return x;
}

// ---------------- Kernel 1: sequential hydrology scan ----------------
// grid: NB_/BASINS_PER_BLOCK = 125 blocks, block: 128 threads.
// thread (bl, m): basin b = blockIdx*16 + bl, multiplier m.
// Params (NB, NT, 96) slab per (block, t) is double-buffered into LDS with
// async b128 copies; P/T/PET are coalesced per-lane streams.
__global__ __launch_bounds__(SCAN_BLOCK) void scan_kernel(
    const float* __restrict__ P, const float* __restrict__ T,
    const float* __restrict__ PET, const float* __restrict__ params,
    float* __restrict__ Q, float* __restrict__ routa, float* __restrict__ routb) {
  __shared__ __align__(16) float prmbuf[2][BASINS_PER_BLOCK * PROW];

  const int tid = threadIdx.x;
  const int bl  = tid >> 3;
  const int m   = tid & 7;
  const int b   = blockIdx.x * BASINS_PER_BLOCK + bl;

  // Per-lane loop-invariant async copy offsets: chunk i covers slab floats
  // [(i*128+tid)*4, +4). Basin-local row padded to PROW in LDS.
  size_t   g_ofs[ASYNC_OPS];
  unsigned l_ofs[ASYNC_OPS];
#pragma unroll
  for (int i = 0; i < ASYNC_OPS; ++i) {
    int e      = (i * SCAN_BLOCK + tid) * 4;   // [0,1536)
    int lb     = e / (NPRM_ * NMUL_);          // local basin 0..15
    int within = e % (NPRM_ * NMUL_);          // 0..92, 16B aligned
    size_t gb  = (size_t)(blockIdx.x * BASINS_PER_BLOCK + lb);
    g_ofs[i] = gb * (size_t)NT_ * (NPRM_ * NMUL_) + (size_t)within;  // float offset
    l_ofs[i] = (unsigned)(lb * PROW + within);
  }

  auto issue = [&](int t, int s) {
#pragma unroll
    for (int i = 0; i < ASYNC_OPS; ++i)
      async_copy_b128(params + g_ofs[i] + (size_t)t * (NPRM_ * NMUL_),
                      &prmbuf[s][l_ofs[i]]);
  };

  issue(0, 0);

  float Ssnow = 0.f, Ssoil = 0.f, Sgw = 0.f;
  const size_t base = (size_t)b * NMUL_ + m;   // per-lane stream offset
  int cur = 0;

  for (int t = 0; t < NT_; ++t) {
    if (t + 1 < NT_) { issue(t + 1, cur ^ 1); wait_asynccnt<ASYNC_OPS>(); }
    else             { wait_asynccnt<0>(); }
    __syncthreads();   // current buffer visible to all waves

    const float* row = &prmbuf[cur][bl * PROW + m];
    // 10 conflict-free LDS reads, then LB/UB affine rescale baked in:
    float tt    = -3.f   + 8.f    * row[0 * 8];
    float ddf   =          20.f   * row[1 * 8];
    float smax  =  1.f   + 1999.f * row[2 * 8];
    float scn   =                   row[3 * 8];
    float scx   =                   row[4 * 8];
    float pexp  =          10.f   * row[5 * 8];
    float krech =          20.f   * row[6 * 8];
    float kif1  =                   row[7 * 8];
    float kif2  =          0.01f  * row[8 * 8];
    float kbf   =                   row[9 * 8];

    const size_t idx = (size_t)t * (NB_ * NMUL_) + base;
    float p  = P[idx];
    float tc = T[idx];
    float pe = PET[idx];
    if (t + 16 < NT_) {  // stream-ahead hints (global_prefetch)
      const size_t f = idx + (size_t)16 * (NB_ * NMUL_);
      __builtin_prefetch(&P[f]);
      __builtin_prefetch(&T[f]);
      __builtin_prefetch(&PET[f]);
    }

    // hydrology step (dt = 1)
    float fsnow    = (tc <= tt) ? 1.0f : 0.0f;
    float snowfall = p * fsnow;
    float rain     = p - snowfall;
    float melt     = fmaxf(fminf(ddf * (tc - tt), Ssnow), 0.0f);
    Ssnow          = Ssnow + snowfall - melt;
    float infil    = rain + melt;
    float invsmax  = 1.0f / smax;
    float qsat     = (scn + (scx - scn) * Ssoil * invsmax) * infil;
    float evap     = fminf(Ssoil * invsmax * pe, Ssoil);
    float rech     = krech * powf(fmaxf(Ssoil, 0.0f) * invsmax + 1e-6f, pexp);
    Ssoil          = fmaxf(Ssoil + infil - qsat - evap - rech, 0.0f);
    float Sc       = fmaxf(Sgw, 0.0f);
    float qif      = fminf(Sc, kif1 * Sc + kif2 * Sc * Sc);
    float qbf      = kbf * Sc;
    Sgw            = fmaxf(Sgw + rech - qif - qbf, 0.0f);

    float qs = grp8_sum(qsat + qif + qbf);
    if (m == 0) Q[(size_t)b * NT_ + t] = qs * 0.125f;  // mean over NMUL

    if (t == NT_ - 1) {  // routing params from the last timestep
      float ra = grp8_sum(3.0f * row[10 * 8]);
      float rb = grp8_sum(5.0f * row[11 * 8]);
      if (m == 0) { routa[b] = ra * 0.125f; routb[b] = rb * 0.125f; }
    }
    __syncthreads();   // everyone done reading buf[cur] before it is refilled
    cur ^= 1;
  }
}

// ---------------- Kernel 2: gamma unit-hydrograph weights ----------------
__global__ void weights_kernel(const float* __restrict__ routa,
                               const float* __restrict__ routb,
                               float* __restrict__ w) {
  int b = blockIdx.x * blockDim.x + threadIdx.x;
  if (b >= NB_) return;
  float aa = fmaxf(routa[b], 0.0f) + 0.1f;
  float th = fmaxf(routb[b], 0.0f) + 0.5f;
  float denom = expf(lgammaf(aa)) * powf(th, aa);
  float vals[LENF_];
  float s = 0.0f;
#pragma unroll
  for (int i = 0; i < LENF_; ++i) {
    float tv = 0.5f + (float)i;
    float v  = powf(tv, aa - 1.0f) * expf(-tv / th) / denom;
    vals[i] = v;
    s += v;
  }
  float inv = 1.0f / s;
#pragma unroll
  for (int i = 0; i < LENF_; ++i) w[b * 16 + i] = vals[i] * inv;
}

// ---------------- Kernel 3: per-basin 15-tap causal FIR ----------------
__global__ void conv_kernel(const float* __restrict__ Q,
                            const float* __restrict__ w,
                            float* __restrict__ out) {
  int idx = blockIdx.x * blockDim.x + threadIdx.x;
  if (idx >= NB_ * NT_) return;
  int b = idx / NT_;
  int t = idx - b * NT_;
  const float* wb = w + b * 16;
  const float* qb = Q + (size_t)b * NT_;
  float acc = 0.0f;
  int kmax = t < (LENF_ - 1) ? t : (LENF_ - 1);
#pragma unroll
  for (int k = 0; k < LENF_; ++k)
    if (k <= kmax) acc = fmaf(wb[k], qb[t - k], acc);
  out[idx] = acc;
}

// ---------------- launch ----------------
extern "C" void kernel_launch(void* const* d_in, const int* in_sizes, int n_in,
                              void* d_out, int out_size, void* d_ws, size_t ws_size,
                              hipStream_t stream) {
  (void)in_sizes; (void)n_in; (void)out_size; (void)ws_size;
  const float* P      = (const float*)d_in[0];
  const float* T      = (const float*)d_in[1];
  const float* PET    = (const float*)d_in[2];
  const float* params = (const float*)d_in[3];
  float* out = (float*)d_out;

  // workspace layout (floats): Q[NB*NT] | routa[NB] | routb[NB] | w[NB*16]
  float* Q     = (float*)d_ws;
  float* routa = Q + (size_t)NB_ * NT_;
  float* routb = routa + NB_;
  float* w     = routb + NB_;

  scan_kernel<<<NB_ / BASINS_PER_BLOCK, SCAN_BLOCK, 0, stream>>>(
      P, T, PET, params, Q, routa, routb);
  weights_kernel<<<(NB_ + 255) / 256, 256, 0, stream>>>(routa, routb, w);
  conv_kernel<<<(NB_ * NT_ + 255) / 256, 256, 0, stream>>>(Q, w, out);
}